// YOLOv1Head_6253472383064
// MI455X (gfx1250) — compile-verified
//
#include <hip/hip_runtime.h>

#define SDIM     7
#define CCLS     20
#define NBOX     98          // 7*7*2
#define CELLS    49
#define IN_PER_B 1470        // 49 * 30
#define OUT_PER_B (NBOX * CCLS * 6)   // 11760 floats = 47040 bytes
#define SCORE_T  0.05f
#define IOU_T    0.5f

typedef unsigned int u32;
typedef u32 v4u __attribute__((ext_vector_type(4)));
typedef int  v8i __attribute__((ext_vector_type(8)));
typedef int  v4i __attribute__((ext_vector_type(4)));

__global__ __launch_bounds__(640)
void yolo_head_kernel(const float* __restrict__ in, float* __restrict__ out) {
  __shared__ float tile[IN_PER_B];                 // raw (7,7,30) tile for this batch elem
  __shared__ float clsp[CELLS * CCLS];             // softmax class probs
  __shared__ float boxpx[NBOX * 4];                // decoded pixel boxes
  __shared__ float svals[CCLS][NBOX];              // per-class scores
  __shared__ unsigned short invp[CCLS][NBOX];      // sorted-pos -> original box idx
  __shared__ __align__(16) float obuf[OUT_PER_B];  // staged output block (47 KB)

  const int    tid    = threadIdx.x;
  const int    blk    = blockIdx.x;
  const size_t inBase = (size_t)blk * IN_PER_B;

  // ---------- Phase 1: TDM async DMA of the input tile into LDS (gfx1250) ----------
  {
    unsigned long long ga = (unsigned long long)(size_t)(const void*)(in + inBase);
    u32 ldsAddr = (u32)(size_t)(void*)tile;        // low 32 bits of generic ptr = LDS byte offset
    v4u g0;
    g0.x = 1u;                                               // count=1, is_restore=0, gather off
    g0.y = ldsAddr;                                          // lds_addr [63:32]
    g0.z = (u32)(ga & 0xffffffffull);                        // global_addr [95:64]
    g0.w = (u32)((ga >> 32) & 0x01ffffffull) | (2u << 30);   // global_addr [120:96], type=2
    v8i g1;
    g1.s0 = 0x00020000;                                      // workgroup_mask=0, data_size=2 (4B)
    g1.s1 = (int)((IN_PER_B & 0xffff) << 16);                // tensor_dim0 lo16
    g1.s2 = (int)(((IN_PER_B >> 16) & 0xffff) | (1 << 16));  // tensor_dim0 hi16 | tensor_dim1=1
    g1.s3 = (int)(IN_PER_B << 16);                           // tile_dim0 = 1470
    g1.s4 = 1;                                               // tile_dim1 = 1, tile_dim2 = 0
    g1.s5 = IN_PER_B;                                        // tensor_dim0_stride lo32
    g1.s6 = (int)((IN_PER_B & 0xffff) << 16);                // stride hi16=0 | dim1_stride lo16
    g1.s7 = (int)(IN_PER_B >> 16);                           // dim1_stride hi bits = 0
    v4i gz  = {0, 0, 0, 0};
    v8i gz8 = {0, 0, 0, 0, 0, 0, 0, 0};
    if (tid < 32) {
      __builtin_amdgcn_tensor_load_to_lds(g0, g1, gz, gz, gz8, 0);
      __builtin_amdgcn_s_wait_tensorcnt(0);
    }
  }
  __syncthreads();
  // Safety net: redundant cooperative load of the same bytes (idempotent).
  for (int t = tid; t < IN_PER_B; t += 640) tile[t] = in[inBase + t];
  __syncthreads();

  // ---------- Phase 2a: per-cell softmax over 20 classes ----------
  if (tid < CELLS) {
    const float* p = &tile[tid * 30];
    float m = p[0];
    for (int c = 1; c < CCLS; ++c) m = fmaxf(m, p[c]);
    float s = 0.f;
    for (int c = 0; c < CCLS; ++c) { float e = expf(p[c] - m); clsp[tid * CCLS + c] = e; s += e; }
    for (int c = 0; c < CCLS; ++c) clsp[tid * CCLS + c] = clsp[tid * CCLS + c] / s;
  }
  // ---------- Phase 2b: decode 98 boxes to clipped pixel coords ----------
  if (tid < NBOX) {
    int cell = tid >> 1, bb = tid & 1;
    int gi = cell / SDIM, gj = cell % SDIM;     // gy=i, gx=j (meshgrid 'ij')
    const float* p = &tile[cell * 30 + CCLS + bb * 4];
    float cx = (p[0] + (float)gj) / 7.0f;
    float cy = (p[1] + (float)gi) / 7.0f;
    float w = p[2], h = p[3];
    float x1 = cx - w * 0.5f, y1 = cy - h * 0.5f;
    float x2 = cx + w * 0.5f, y2 = cy + h * 0.5f;
    boxpx[tid * 4 + 0] = fminf(fmaxf(x1, 0.f), 1.f) * 448.f;
    boxpx[tid * 4 + 1] = fminf(fmaxf(y1, 0.f), 1.f) * 448.f;
    boxpx[tid * 4 + 2] = fminf(fmaxf(x2, 0.f), 1.f) * 448.f;
    boxpx[tid * 4 + 3] = fminf(fmaxf(y2, 0.f), 1.f) * 448.f;
  }
  __syncthreads();
  // ---------- Phase 2c: scores[n,c] = cls_prob[cell,c] * conf[cell,bb] ----------
  for (int e = tid; e < NBOX * CCLS; e += 640) {
    int n = e / CCLS, c = e - n * CCLS;
    int cell = n >> 1, bb = n & 1;
    svals[c][n] = clsp[cell * CCLS + c] * tile[cell * 30 + 28 + bb];
  }
  __syncthreads();

  // ---------- Phase 3: one wave per class: stable descending rank + greedy NMS ----------
  const int lane = tid & 31;
  const int c    = tid >> 5;                     // 0..19
  const float* sc = svals[c];

  // Stable rank = #{ s[j] > s[i] } + #{ s[j]==s[i], j<i }  (matches jnp.argsort(-s))
  #pragma unroll
  for (int w = 0; w < 4; ++w) {
    int i = lane + 32 * w;
    if (i < NBOX) {
      float si = sc[i];
      int r = 0;
      for (int j = 0; j < NBOX; ++j) {
        float sj = sc[j];
        r += (sj > si) || ((sj == si) && (j < i));
      }
      invp[c][r] = (unsigned short)i;
    }
  }
  __syncthreads();

  // Each lane owns sorted positions p = lane + 32*w, keep bits in kmask (register).
  int   bi[4];
  float bx1[4], by1[4], bx2[4], by2[4], ar[4];
  u32   kmask = 0;
  #pragma unroll
  for (int w = 0; w < 4; ++w) {
    int p = lane + 32 * w;
    if (p < NBOX) {
      int n = invp[c][p];
      bi[w]  = n;
      bx1[w] = boxpx[n * 4 + 0]; by1[w] = boxpx[n * 4 + 1];
      bx2[w] = boxpx[n * 4 + 2]; by2[w] = boxpx[n * 4 + 3];
      ar[w]  = fmaxf(bx2[w] - bx1[w], 0.f) * fmaxf(by2[w] - by1[w], 0.f);
      kmask |= (1u << w);
    } else {
      bi[w] = 0; bx1[w] = by1[w] = bx2[w] = by2[w] = ar[w] = 0.f;
    }
  }

  // Greedy suppression: keep[i] broadcast via wave32 ballot; box i is read from LDS.
  for (int i = 0; i < NBOX; ++i) {
    int wi = i >> 5, b = i & 31;
    u32 m = __builtin_amdgcn_ballot_w32((kmask >> wi) & 1u);
    if ((m >> b) & 1u) {
      int oi = invp[c][i];
      float ix1 = boxpx[oi * 4 + 0], iy1 = boxpx[oi * 4 + 1];
      float ix2 = boxpx[oi * 4 + 2], iy2 = boxpx[oi * 4 + 3];
      float ai  = fmaxf(ix2 - ix1, 0.f) * fmaxf(iy2 - iy1, 0.f);
      #pragma unroll
      for (int w2 = 0; w2 < 4; ++w2) {
        int p = lane + 32 * w2;
        if (p > i && ((kmask >> w2) & 1u)) {
          float xx1 = fmaxf(ix1, bx1[w2]);
          float yy1 = fmaxf(iy1, by1[w2]);
          float xx2 = fminf(ix2, bx2[w2]);
          float yy2 = fminf(iy2, by2[w2]);
          float inter = fmaxf(xx2 - xx1, 0.f) * fmaxf(yy2 - yy1, 0.f);
          float uni   = ai + ar[w2] - inter;
          float iou   = inter / fmaxf(uni, 1e-9f);
          if (iou > IOU_T) kmask &= ~(1u << w2);
        }
      }
    }
  }

  // ---------- Phase 4: masked rows -> LDS staging buffer (scattered writes are free) ----------
  #pragma unroll
  for (int w2 = 0; w2 < 4; ++w2) {
    int p = lane + 32 * w2;
    if (p < NBOX) {
      int n = bi[w2];
      float sp = sc[n];
      bool kp = ((kmask >> w2) & 1u) && (sp >= SCORE_T);
      float* r = &obuf[((u32)n * CCLS + (u32)c) * 6];
      if (kp) {
        r[0] = (float)c;
        r[1] = bx1[w2]; r[2] = by1[w2]; r[3] = bx2[w2]; r[4] = by2[w2];
        r[5] = sp;
      } else {
        r[0] = 0.f; r[1] = 0.f; r[2] = 0.f; r[3] = 0.f; r[4] = 0.f; r[5] = 0.f;
      }
    }
  }
  __syncthreads();

  // ---------- Phase 5: coalesced streaming store of the whole block (b128) ----------
  {
    const float4* src = (const float4*)obuf;
    float4* dst = (float4*)(out + (size_t)blk * OUT_PER_B);   // 47040 B/block, 16B aligned
    #pragma unroll
    for (int t = tid; t < OUT_PER_B / 4; t += 640) dst[t] = src[t];
  }
}

extern "C" void kernel_launch(void* const* d_in, const int* in_sizes, int n_in,
                              void* d_out, int out_size, void* d_ws, size_t ws_size,
                              hipStream_t stream) {
  const float* in  = (const float*)d_in[0];
  float*       out = (float*)d_out;
  int Bn = in_sizes[0] / IN_PER_B;   // 1024 for the reference setup
  dim3 grid(Bn), block(640);
  hipLaunchKernelGGL(yolo_head_kernel, grid, block, 0, stream, in, out);
}